// EdgeConvBlock_63702954934706
// MI455X (gfx1250) — compile-verified
//
#include <hip/hip_runtime.h>

typedef __attribute__((ext_vector_type(2))) float v2f;
typedef __attribute__((ext_vector_type(8))) float v8f;

constexpr int BB = 16;     // batch
constexpr int NN = 2048;   // points
constexpr int KK = 32;     // neighbors

__device__ __forceinline__ v8f wmma_f32(v2f a, v2f b, v8f c) {
  // V_WMMA_F32_16X16X4_F32: D(16x16,f32) = A(16x4,f32) * B(4x16,f32) + C
  return __builtin_amdgcn_wmma_f32_16x16x4_f32(
      /*neg_a=*/false, a, /*neg_b=*/false, b,
      /*c_mod=*/(short)0, c, /*reuse_a=*/false, /*reuse_b=*/false);
}

// Padded LDS index: +1 float per 32 -> strided scans (33/66-float strides)
// hit distinct banks (gcd(33,64)=1; 66 -> even offsets all distinct mod 64).
__device__ __forceinline__ int dind(int m) { return m + (m >> 5); }

// ---------------------------------------------------------------------------
// Squared norms: sq[b*N+n] = sum_c h[b][c][n]^2
// ---------------------------------------------------------------------------
template<int C>
__global__ __launch_bounds__(256) void sq_kernel(const float* __restrict__ h,
                                                 float* __restrict__ sq) {
  int i = blockIdx.x * 256 + threadIdx.x;           // i = b*N + n
  int b = i >> 11, n = i & (NN - 1);
  const float* p = h + (size_t)b * C * NN + n;
  float s = 0.f;
#pragma unroll
  for (int c = 0; c < C; ++c) {
    float v = p[(size_t)c * NN];
    s = fmaf(v, v, s);
  }
  sq[i] = s;
}

// ---------------------------------------------------------------------------
// Shared top-32 selection over one padded LDS distance row (whole wave).
// Returns this lane's k-th index (k = lane), descending, ties -> low index.
// ---------------------------------------------------------------------------
__device__ __forceinline__ int topk_row(float* drow, int lane) {
  // per-lane cached argmax over its 64 strided elements (m = lane + 32*i)
  float best = -3.4e38f;
  int besti = 0x7fffffff;
#pragma unroll 8
  for (int i = 0; i < NN / 32; ++i) {
    int m = lane + 32 * i;
    float v = drow[dind(m)];
    if (v > best || (v == best && m < besti)) { best = v; besti = m; }
  }
  int myidx = 0;
  for (int k = 0; k < KK; ++k) {
    float gb = best;
    int gi = besti;
#pragma unroll
    for (int off = 1; off < 32; off <<= 1) {
      float ov = __shfl_xor(gb, off, 32);
      int   oi = __shfl_xor(gi, off, 32);
      if (ov > gb || (ov == gb && oi < gi)) { gb = ov; gi = oi; }
    }
    if (lane == k) myidx = gi;
    if (k == KK - 1) break;

    const int o = gi & 31;                          // owner lane of removed elem
    if (lane == 0) drow[dind(gi)] = -3.4e38f;
    asm volatile("s_wait_dscnt 0" ::: "memory");

    // cooperative re-scan of owner's 64 elements (2 per lane)
    float nb = -3.4e38f;
    int ni = 0x7fffffff;
#pragma unroll
    for (int t = 0; t < 2; ++t) {
      int m = o + 32 * (2 * lane + t);
      float v = drow[dind(m)];
      if (v > nb || (v == nb && m < ni)) { nb = v; ni = m; }
    }
#pragma unroll
    for (int off = 1; off < 32; off <<= 1) {
      float ov = __shfl_xor(nb, off, 32);
      int   oi = __shfl_xor(ni, off, 32);
      if (ov > nb || (ov == nb && oi < ni)) { nb = ov; ni = oi; }
    }
    if (lane == o) { best = nb; besti = ni; }
  }
  return myidx;
}

// ---------------------------------------------------------------------------
// KNN (C=3, layer 0): scalar distances, one wave per row.
// ---------------------------------------------------------------------------
template<int C>
__global__ __launch_bounds__(128) void knn_kernel(const float* __restrict__ h,
                                                  const float* __restrict__ sq,
                                                  int* __restrict__ idx) {
  __shared__ float dist[4][NN + NN / 32];           // 2112 floats / wave
  const int lane = threadIdx.x & 31;
  const int wave = threadIdx.x >> 5;
  const int gid  = blockIdx.x * 4 + wave;           // b*N + n
  const int b = gid >> 11;
  const int n = gid & (NN - 1);
  const float* hb = h + (size_t)b * C * NN;
  float* drow = dist[wave];

  float xn[C];
#pragma unroll
  for (int c = 0; c < C; ++c) xn[c] = hb[(size_t)c * NN + n];
  const float sqn = sq[gid];

  for (int m0 = 0; m0 < NN; m0 += 32) {
    int m = m0 + lane;
    float dot = 0.f;
#pragma unroll
    for (int c = 0; c < C; ++c) dot = fmaf(xn[c], hb[(size_t)c * NN + m], dot);
    drow[dind(m)] = 2.f * dot - sqn - sq[b * NN + m];
  }
  asm volatile("s_wait_dscnt 0" ::: "memory");

  int myidx = topk_row(drow, lane);
  idx[(size_t)gid * KK + lane] = myidx;
}

// ---------------------------------------------------------------------------
// KNN (C=64): WMMA gram. Workgroup = 4 waves = 16 rows of one batch.
// Phase 1: each wave computes 4 col-chunks of 128 for all 16 rows via
//          v_wmma_f32_16x16x4_f32 into a shared 16 x 2048 LDS dist block.
// Phase 2: each wave runs top-32 on 4 rows.
// Row stride 2116 floats: half-wave epilogue stores land 32 banks apart.
// ---------------------------------------------------------------------------
__global__ __launch_bounds__(128) void knn_wmma_kernel(const float* __restrict__ h,
                                                       const float* __restrict__ sq,
                                                       int* __restrict__ idx) {
  constexpr int C = 64;
  constexpr int DST = 2116;                         // padded row stride (floats)
  __shared__ float dist[16 * DST];                  // ~132 KB (<= 320 KB/WG)
  const int lane = threadIdx.x & 31;
  const int wave = threadIdx.x >> 5;
  const int rowbase = blockIdx.x * 16;              // linear b*N + nbase
  const int b = rowbase >> 11;
  const int nbase = rowbase & (NN - 1);
  const float* hb = h + (size_t)b * C * NN;
  const float* sqb = sq + b * NN;

  const int half = lane >> 4;
  const int l16  = lane & 15;

  // Preload A fragments: 16 rows (points) x 64 channels, 16 k-steps
  v2f af[16];
#pragma unroll
  for (int ks = 0; ks < 16; ++ks) {
    const int c0 = 4 * ks + 2 * half;
    af[ks].x = hb[(size_t)c0 * NN + nbase + l16];
    af[ks].y = hb[(size_t)(c0 + 1) * NN + nbase + l16];
  }
  float sqr[8];
#pragma unroll
  for (int r = 0; r < 8; ++r) sqr[r] = sqb[nbase + r + 8 * half];

  // ---- Phase 1: gram -> neg dist (4 chunks of 128 cols per wave) ----
  for (int ch = 0; ch < 4; ++ch) {
    const int colbase = (wave * 4 + ch) * 128;
    v8f acc[8];
#pragma unroll
    for (int nt = 0; nt < 8; ++nt) acc[nt] = (v8f){};
#pragma unroll
    for (int ks = 0; ks < 16; ++ks) {
      const int c0 = 4 * ks + 2 * half;
      const float* r0 = hb + (size_t)c0 * NN + colbase + l16;
      const float* r1 = hb + (size_t)(c0 + 1) * NN + colbase + l16;
#pragma unroll
      for (int nt = 0; nt < 8; ++nt) {
        v2f bf;
        bf.x = r0[16 * nt];
        bf.y = r1[16 * nt];
        acc[nt] = wmma_f32(af[ks], bf, acc[nt]);
      }
    }
#pragma unroll
    for (int nt = 0; nt < 8; ++nt) {
      const int col = colbase + 16 * nt + l16;
      const float sqc = sqb[col];
#pragma unroll
      for (int r = 0; r < 8; ++r) {
        const int M = r + 8 * half;                 // row within 16-row block
        dist[M * DST + dind(col)] = 2.f * acc[nt][r] - sqr[r] - sqc;
      }
    }
  }
  __syncthreads();

  // ---- Phase 2: top-32 per row; wave handles rows 4w..4w+3 ----
  for (int rr = 0; rr < 4; ++rr) {
    const int M = wave * 4 + rr;
    int myidx = topk_row(dist + M * DST, lane);
    idx[(size_t)(rowbase + M) * KK + lane] = myidx;
  }
}

// ---------------------------------------------------------------------------
// Fused EdgeConv MLP: one wave per point (unchanged from round 2).
// ---------------------------------------------------------------------------
template<int CIN, int KS1>
__global__ __launch_bounds__(128) void conv_kernel(
    const float* __restrict__ h, const int* __restrict__ idx,
    const float* __restrict__ w1, const float* __restrict__ s1, const float* __restrict__ t1,
    const float* __restrict__ w2, const float* __restrict__ s2, const float* __restrict__ t2,
    float* __restrict__ out_concat, float* __restrict__ out_x, int layer) {
  __shared__ float Zs[4][64 * KK];                  // 32 KB
  const int lane = threadIdx.x & 31;
  const int wave = threadIdx.x >> 5;
  const int gid  = blockIdx.x * 4 + wave;           // b*N + n
  const int b = gid >> 11;
  const int n = gid & (NN - 1);
  const float* hb = h + (size_t)b * CIN * NN;
  float* Zw = Zs[wave];

  __builtin_prefetch(w1, 0, 3);                     // global_prefetch_b8
  __builtin_prefetch(w2, 0, 3);

  const int half = lane >> 4;                       // 0 / 1
  const int l16  = lane & 15;
  const int ji0  = idx[(size_t)gid * KK + l16];       // col  0..15
  const int ji1  = idx[(size_t)gid * KK + l16 + 16];  // col 16..31

  auto felem = [&](int c, int jidx) -> float {      // generic (CIN=3 only)
    if (c < CIN) return hb[(size_t)c * NN + n];                       // center
    if (c < 2 * CIN) {
      int cc = c - CIN;
      return hb[(size_t)cc * NN + jidx] - hb[(size_t)cc * NN + n];    // nbr-center
    }
    return 0.f;                                                       // K-pad
  };

  v8f acc[4][2];
#pragma unroll
  for (int mt = 0; mt < 4; ++mt)
#pragma unroll
    for (int nt = 0; nt < 2; ++nt) acc[mt][nt] = (v8f){};

  // ---- conv1: Y1(64x32) = W1(64x2C) * F(2Cx32) ----
#pragma unroll
  for (int ks = 0; ks < KS1; ++ks) {
    const int c0 = 4 * ks;
    const int ra = c0 + 2 * half, rb = ra + 1;      // B rows per ISA f32 layout
    v2f bf0, bf1;
    if constexpr (CIN == 64) {
      if (ks < KS1 / 2) {
        // center rows: independent of neighbor index -> bf0 == bf1
        bf0.x = hb[(size_t)ra * NN + n];
        bf0.y = hb[(size_t)rb * NN + n];
        bf1 = bf0;
      } else {
        const int ca = ra - CIN, cb = ca + 1;
        const float cena = hb[(size_t)ca * NN + n];
        const float cenb = hb[(size_t)cb * NN + n];
        bf0.x = hb[(size_t)ca * NN + ji0] - cena;
        bf0.y = hb[(size_t)cb * NN + ji0] - cenb;
        bf1.x = hb[(size_t)ca * NN + ji1] - cena;
        bf1.y = hb[(size_t)cb * NN + ji1] - cenb;
      }
    } else {
      bf0.x = felem(ra, ji0); bf0.y = felem(rb, ji0);
      bf1.x = felem(ra, ji1); bf1.y = felem(rb, ji1);
    }
#pragma unroll
    for (int mt = 0; mt < 4; ++mt) {
      const int row = 16 * mt + l16;
      const int ca = c0 + 2 * half, cb = ca + 1;    // A cols per ISA f32 layout
      v2f af;
      if constexpr ((2 * CIN) % 8 == 0) {           // no K padding needed
        af.x = w1[(size_t)row * (2 * CIN) + ca];
        af.y = w1[(size_t)row * (2 * CIN) + cb];
      } else {
        af.x = (ca < 2 * CIN) ? w1[(size_t)row * (2 * CIN) + ca] : 0.f;
        af.y = (cb < 2 * CIN) ? w1[(size_t)row * (2 * CIN) + cb] : 0.f;
      }
      acc[mt][0] = wmma_f32(af, bf0, acc[mt][0]);
      acc[mt][1] = wmma_f32(af, bf1, acc[mt][1]);
    }
  }

  // ---- BN + LeakyReLU -> Z (LDS); D layout: elem r -> M = 16*mt + r + 8*half
#pragma unroll
  for (int mt = 0; mt < 4; ++mt) {
#pragma unroll
    for (int r = 0; r < 8; ++r) {
      const int M = 16 * mt + r + 8 * half;
      const float sc = s1[M], sh = t1[M];
#pragma unroll
      for (int nt = 0; nt < 2; ++nt) {
        const int col = 16 * nt + l16;
        float z = fmaf(acc[mt][nt][r], sc, sh);
        z = z > 0.f ? z : 0.2f * z;
        Zw[M * KK + col] = z;
      }
    }
  }
  asm volatile("s_wait_dscnt 0" ::: "memory");

  // ---- conv2: Y2(64x32) = W2(64x64) * Z(64x32) ----
  v8f acc2[4][2];
#pragma unroll
  for (int mt = 0; mt < 4; ++mt)
#pragma unroll
    for (int nt = 0; nt < 2; ++nt) acc2[mt][nt] = (v8f){};

#pragma unroll
  for (int ks = 0; ks < 16; ++ks) {
    const int c0 = 4 * ks;
    const int ra = c0 + 2 * half;
    v2f bf0, bf1;
    bf0.x = Zw[ra * KK + l16];        bf0.y = Zw[(ra + 1) * KK + l16];
    bf1.x = Zw[ra * KK + 16 + l16];   bf1.y = Zw[(ra + 1) * KK + 16 + l16];
#pragma unroll
    for (int mt = 0; mt < 4; ++mt) {
      const int row = 16 * mt + l16;
      v2f af;
      af.x = w2[(size_t)row * 64 + ra];
      af.y = w2[(size_t)row * 64 + ra + 1];
      acc2[mt][0] = wmma_f32(af, bf0, acc2[mt][0]);
      acc2[mt][1] = wmma_f32(af, bf1, acc2[mt][1]);
    }
  }

  // ---- BN + LeakyReLU + max over 32 neighbors, write both output regions
#pragma unroll
  for (int mt = 0; mt < 4; ++mt) {
#pragma unroll
    for (int r = 0; r < 8; ++r) {
      const int M = 16 * mt + r + 8 * half;
      const float sc = s2[M], sh = t2[M];
      float za = fmaf(acc2[mt][0][r], sc, sh); za = za > 0.f ? za : 0.2f * za;
      float zb = fmaf(acc2[mt][1][r], sc, sh); zb = zb > 0.f ? zb : 0.2f * zb;
      float m = fmaxf(za, zb);
#pragma unroll
      for (int off = 1; off < 16; off <<= 1) m = fmaxf(m, __shfl_xor(m, off, 32));
      if (l16 == r) {
        out_concat[((size_t)b * 192 + layer * 64 + M) * NN + n] = m;
        out_x[((size_t)b * 64 + M) * NN + n] = m;
      }
    }
  }
}

// ---------------------------------------------------------------------------
extern "C" void kernel_launch(void* const* d_in, const int* in_sizes, int n_in,
                              void* d_out, int out_size, void* d_ws, size_t ws_size,
                              hipStream_t stream) {
  const float* x = (const float*)d_in[0];           // (B,3,N); d_in[1] coordinate unused
  float* out  = (float*)d_out;                      // (B,192,N) concat
  float* xreg = out + (size_t)BB * 192 * NN;        // x0,x1,x2 regions
  float* sq   = (float*)d_ws;                       // B*N floats
  int*   idx  = (int*)(sq + (size_t)BB * NN);       // B*N*K ints

  const int BN = BB * NN;                           // 32768
  for (int l = 0; l < 3; ++l) {
    const float* h  = (l == 0) ? x : (const float*)(xreg + (size_t)(l - 1) * BB * 64 * NN);
    const float* w1 = (const float*)d_in[2 + 6 * l + 0];
    const float* s1 = (const float*)d_in[2 + 6 * l + 1];
    const float* t1 = (const float*)d_in[2 + 6 * l + 2];
    const float* w2 = (const float*)d_in[2 + 6 * l + 3];
    const float* s2 = (const float*)d_in[2 + 6 * l + 4];
    const float* t2 = (const float*)d_in[2 + 6 * l + 5];
    float* ox = xreg + (size_t)l * BB * 64 * NN;
    if (l == 0) {
      sq_kernel<3><<<BN / 256, 256, 0, stream>>>(h, sq);
      knn_kernel<3><<<BN / 4, 128, 0, stream>>>(h, sq, idx);
      conv_kernel<3, 2><<<BN / 4, 128, 0, stream>>>(h, idx, w1, s1, t1, w2, s2, t2, out, ox, l);
    } else {
      sq_kernel<64><<<BN / 256, 256, 0, stream>>>(h, sq);
      knn_wmma_kernel<<<BN / 16, 128, 0, stream>>>(h, sq, idx);
      conv_kernel<64, 32><<<BN / 4, 128, 0, stream>>>(h, idx, w1, s1, t1, w2, s2, t2, out, ox, l);
    }
  }
}